// EdgeConv_39419209842979
// MI455X (gfx1250) — compile-verified
//
#include <hip/hip_runtime.h>
#include <hip/hip_bf16.h>

#define NN 50000
#define NE 1600000

typedef __attribute__((ext_vector_type(2))) float v2f;
typedef __attribute__((ext_vector_type(8))) float v8f;

// monotone float <-> uint mapping for atomicMax on floats (handles negatives, -inf init)
__device__ __forceinline__ unsigned f2ord(float f) {
  unsigned u = __float_as_uint(f);
  return (u & 0x80000000u) ? ~u : (u | 0x80000000u);
}
__device__ __forceinline__ float ord2f(unsigned u) {
  return __uint_as_float((u & 0x80000000u) ? (u & 0x7fffffffu) : ~u);
}

struct Params {
  const float *x, *en, *D_en;
  const int *src, *dst;
  const float *theta_w, *theta_b, *phi_w, *phi_b;
  const float *t_[8];   // theta_en mlp leaves: W0,b0,W1,b1,W2,b2,W3,b3
  const float *p_[8];   // phi_en mlp leaves
  const float *w_[8];   // score mlp leaves
  const float *TW0, *PW0, *hb;   // fused (in workspace)
  float *x_out, *en_out, *sc_out, *deg;
};

// 8 -> 4 -> 5 -> 4 -> 8 MLP (relu hidden, linear out), row-major weights
__device__ __forceinline__ void mlp_8_8(const float in[8], const float* const W[8], float out[8]) {
  float h0[4];
#pragma unroll
  for (int j = 0; j < 4; ++j) {
    float a = W[1][j];
#pragma unroll
    for (int k = 0; k < 8; ++k) a += in[k] * W[0][k * 4 + j];
    h0[j] = fmaxf(a, 0.f);
  }
  float h1[5];
#pragma unroll
  for (int j = 0; j < 5; ++j) {
    float a = W[3][j];
#pragma unroll
    for (int k = 0; k < 4; ++k) a += h0[k] * W[2][k * 5 + j];
    h1[j] = fmaxf(a, 0.f);
  }
  float h2[4];
#pragma unroll
  for (int j = 0; j < 4; ++j) {
    float a = W[5][j];
#pragma unroll
    for (int k = 0; k < 5; ++k) a += h1[k] * W[4][k * 4 + j];
    h2[j] = fmaxf(a, 0.f);
  }
#pragma unroll
  for (int j = 0; j < 8; ++j) {
    float a = W[7][j];
#pragma unroll
    for (int k = 0; k < 4; ++k) a += h2[k] * W[6][k * 8 + j];
    out[j] = a;
  }
}

// ---------------- init: x region = ord(-inf); en/score regions = 0; deg = 0 ----------------
__global__ __launch_bounds__(256) void init_kernel(float* out, float* deg) {
  int i = blockIdx.x * blockDim.x + threadIdx.x;
  if (i < NN * 32) {
    ((unsigned*)out)[i] = 0x007FFFFFu;   // f2ord(-inf)
  } else if (i < NN * 41) {
    out[i] = 0.f;
  }
  if (i < NN) deg[i] = 0.f;
}

// ---------------- precompute fused score-head matrices (once per launch) ----------------
// f[0..127]   = TW0[k][j] = sum_n theta_w[k][n] * W0[n][j]
// f[128..255] = PW0[k][j] = sum_n phi_w[k][n]   * W0[32+n][j]
// f[256..259] = hb[j]     = b0[j] + theta_b@W0[0:32] + phi_b@W0[32:64]
__global__ void fuse_kernel(const float* tw, const float* tb, const float* pw, const float* pb,
                            const float* wW0, const float* wb0, float* f) {
  const int t = threadIdx.x;   // 128 threads
  const int k = t >> 2, j = t & 3;
  float aT = 0.f, aP = 0.f;
  for (int n = 0; n < 32; ++n) {
    aT += tw[k * 32 + n] * wW0[n * 4 + j];
    aP += pw[k * 32 + n] * wW0[(32 + n) * 4 + j];
  }
  f[t] = aT;
  f[128 + t] = aP;
  if (t < 4) {
    float a = wb0[t];
    for (int n = 0; n < 32; ++n) a += tb[n] * wW0[n * 4 + t] + pb[n] * wW0[(32 + n) * 4 + t];
    f[256 + t] = a;
  }
}

// ---------------- main edge kernel: one wave32 per 16-edge tile ----------------
__global__ __launch_bounds__(256) void edge_kernel(Params p) {
  const int wave = threadIdx.x >> 5;
  const int lane = threadIdx.x & 31;
  const int tile = blockIdx.x * 8 + wave;       // NE/16 = 100000 tiles exactly
  const int row  = lane & 15;
  const int hi   = lane >> 4;
  const int e    = tile * 16 + row;

  const int s = p.src[e];
  const int d = p.dst[e];

  // ---- WMMA path: edge_x[16,32] = (xd-xs)@theta_w + xs@phi_w + biases ----
  // A layout (f32 16x4): lane = M row (0..15), VGPR0/1 = K = 4*kb + 2*hi + {0,1}
  v2f at[8], ap[8];
  {
    const float* xs = p.x + (size_t)s * 32 + 2 * hi;
    const float* xd = p.x + (size_t)d * 32 + 2 * hi;
#pragma unroll
    for (int kb = 0; kb < 8; ++kb) {
      v2f a; a.x = xs[kb * 4]; a.y = xs[kb * 4 + 1];
      v2f b; b.x = xd[kb * 4]; b.y = xd[kb * 4 + 1];
      ap[kb] = a;
      at[kb] = b - a;
    }
  }
  v8f ct0 = {0,0,0,0,0,0,0,0}, ct1 = {0,0,0,0,0,0,0,0};
  v8f cp0 = {0,0,0,0,0,0,0,0}, cp1 = {0,0,0,0,0,0,0,0};
#pragma unroll
  for (int kb = 0; kb < 8; ++kb) {
    const int k0 = kb * 4 + 2 * hi;   // B layout: lane = N col, VGPR0/1 = K rows k0, k0+1
    v2f bt0, bt1, bp0, bp1;
    bt0.x = p.theta_w[k0 * 32 + row];            bt0.y = p.theta_w[(k0 + 1) * 32 + row];
    bt1.x = p.theta_w[k0 * 32 + 16 + row];       bt1.y = p.theta_w[(k0 + 1) * 32 + 16 + row];
    bp0.x = p.phi_w[k0 * 32 + row];              bp0.y = p.phi_w[(k0 + 1) * 32 + row];
    bp1.x = p.phi_w[k0 * 32 + 16 + row];         bp1.y = p.phi_w[(k0 + 1) * 32 + 16 + row];
    ct0 = __builtin_amdgcn_wmma_f32_16x16x4_f32(false, at[kb], false, bt0, (short)0, ct0, false, false);
    ct1 = __builtin_amdgcn_wmma_f32_16x16x4_f32(false, at[kb], false, bt1, (short)0, ct1, false, false);
    cp0 = __builtin_amdgcn_wmma_f32_16x16x4_f32(false, ap[kb], false, bp0, (short)0, cp0, false, false);
    cp1 = __builtin_amdgcn_wmma_f32_16x16x4_f32(false, ap[kb], false, bp1, (short)0, cp1, false, false);
  }
  // scatter-max edge_x into x_out. D layout: lane = N col, VGPR r = M row (r + 8*hi)
  {
    const float bias0 = p.theta_b[row]      + p.phi_b[row];
    const float bias1 = p.theta_b[16 + row] + p.phi_b[16 + row];
    unsigned* xo = (unsigned*)p.x_out;
#pragma unroll
    for (int r = 0; r < 8; ++r) {
      const int m = r + 8 * hi;
      const int node = p.dst[tile * 16 + m];
      atomicMax(xo + (size_t)node * 32 + row,      f2ord(ct0[r] + cp0[r] + bias0));
      atomicMax(xo + (size_t)node * 32 + 16 + row, f2ord(ct1[r] + cp1[r] + bias1));
    }
  }

  // ---- scalar path: lanes 0..15 each own one edge (tiny MLPs + fused score head) ----
  if (hi == 0) {
    float es[8], tin[8];
#pragma unroll
    for (int j = 0; j < 8; ++j) {
      es[j]  = p.en[(size_t)s * 8 + j];
      tin[j] = p.en[(size_t)d * 8 + j] - es[j];
    }
    float tm[8], pm[8];
    mlp_8_8(tin, p.t_, tm);
    mlp_8_8(es,  p.p_, pm);
#pragma unroll
    for (int j = 0; j < 8; ++j)
      atomicAdd(p.en_out + (size_t)d * 8 + j, tm[j] + pm[j]);

    // score head: h1 = hb + (xd-xs)@TW0 + xs@PW0 + en-terms + D_en-terms (W0 bias folded in hb)
    float h1[4];
#pragma unroll
    for (int j = 0; j < 4; ++j) h1[j] = p.hb[j];
    for (int k = 0; k < 32; ++k) {
      const float xsv = p.x[(size_t)s * 32 + k];
      const float dxv = p.x[(size_t)d * 32 + k] - xsv;
#pragma unroll
      for (int j = 0; j < 4; ++j)
        h1[j] += dxv * p.TW0[k * 4 + j] + xsv * p.PW0[k * 4 + j];
    }
#pragma unroll
    for (int i = 0; i < 8; ++i)
#pragma unroll
      for (int j = 0; j < 4; ++j)
        h1[j] += tm[i] * p.w_[0][(64 + i) * 4 + j] + pm[i] * p.w_[0][(72 + i) * 4 + j];
#pragma unroll
    for (int c = 0; c < 3; ++c) {
      const float dv = p.D_en[(size_t)e * 3 + c];
#pragma unroll
      for (int j = 0; j < 4; ++j) h1[j] += dv * p.w_[0][(80 + c) * 4 + j];
    }
#pragma unroll
    for (int j = 0; j < 4; ++j) h1[j] = fmaxf(h1[j], 0.f);

    float h2[5];
#pragma unroll
    for (int j = 0; j < 5; ++j) {
      float a = p.w_[3][j];
#pragma unroll
      for (int k = 0; k < 4; ++k) a += h1[k] * p.w_[2][k * 5 + j];
      h2[j] = fmaxf(a, 0.f);
    }
    float h3[4];
#pragma unroll
    for (int j = 0; j < 4; ++j) {
      float a = p.w_[5][j];
#pragma unroll
      for (int k = 0; k < 5; ++k) a += h2[k] * p.w_[4][k * 4 + j];
      h3[j] = fmaxf(a, 0.f);
    }
    float z = p.w_[7][0];
#pragma unroll
    for (int k = 0; k < 4; ++k) z += h3[k] * p.w_[6][k];
    const float sc = 1.f / (1.f + expf(-z));
    atomicAdd(p.sc_out + d, sc);
    atomicAdd(p.deg + d, 1.f);
  }
}

// ---------------- finalize: decode max-encoding, mask zero-degree, divide means ----------------
__global__ __launch_bounds__(256) void final_kernel(float* out, const float* deg) {
  int i = blockIdx.x * blockDim.x + threadIdx.x;
  if (i >= NN * 41) return;
  if (i < NN * 32) {
    const int node = i >> 5;
    const unsigned u = ((const unsigned*)out)[i];
    out[i] = (deg[node] > 0.f) ? ord2f(u) : 0.f;
  } else if (i < NN * 40) {
    const int node = (i - NN * 32) >> 3;
    out[i] = out[i] / fmaxf(deg[node], 1.f);
  } else {
    const int node = i - NN * 40;
    out[i] = out[i] / fmaxf(deg[node], 1.f);
  }
}

extern "C" void kernel_launch(void* const* d_in, const int* in_sizes, int n_in,
                              void* d_out, int out_size, void* d_ws, size_t ws_size,
                              hipStream_t stream) {
  Params p;
  p.x       = (const float*)d_in[0];
  p.en      = (const float*)d_in[1];
  p.D_en    = (const float*)d_in[2];
  p.src     = (const int*)d_in[3];
  p.dst     = (const int*)d_in[4];
  p.theta_w = (const float*)d_in[5];
  p.theta_b = (const float*)d_in[6];
  p.phi_w   = (const float*)d_in[7];
  p.phi_b   = (const float*)d_in[8];

  if (n_in >= 33) {
    // pytree leaves flattened into individual inputs
    for (int i = 0; i < 8; ++i) {
      p.t_[i] = (const float*)d_in[9 + i];
      p.p_[i] = (const float*)d_in[17 + i];
      p.w_[i] = (const float*)d_in[25 + i];
    }
  } else {
    // each param list is one concatenated blob: W0,b0,W1,b1,W2,b2,W3,b3
    const float* t  = (const float*)d_in[9];
    const float* pp = (const float*)d_in[10];
    const float* w  = (const float*)d_in[11];
    static const int off8[8] = {0, 32, 36, 56, 61, 81, 85, 117};     // [8,4,5,4,8] mlp
    static const int offw[8] = {0, 332, 336, 356, 361, 381, 385, 389}; // [83,4,5,4,1] mlp
    for (int i = 0; i < 8; ++i) {
      p.t_[i] = t + off8[i];
      p.p_[i] = pp + off8[i];
      p.w_[i] = w + offw[i];
    }
  }

  // workspace: deg[NN] | TW0[128] | PW0[128] | hb[4]
  float* ws = (float*)d_ws;
  float* deg  = ws;
  float* fuse = ws + NN;
  p.deg = deg;
  p.TW0 = fuse;
  p.PW0 = fuse + 128;
  p.hb  = fuse + 256;

  p.x_out  = (float*)d_out;
  p.en_out = p.x_out + (size_t)NN * 32;
  p.sc_out = p.en_out + (size_t)NN * 8;

  const int tot = NN * 41;
  init_kernel<<<(tot + 255) / 256, 256, 0, stream>>>((float*)d_out, deg);
  fuse_kernel<<<1, 128, 0, stream>>>(p.theta_w, p.theta_b, p.phi_w, p.phi_b,
                                     p.w_[0], p.w_[1], fuse);
  edge_kernel<<<NE / 16 / 8, 256, 0, stream>>>(p);   // 12500 blocks x 8 waves, 1 tile/wave
  final_kernel<<<(tot + 255) / 256, 256, 0, stream>>>((float*)d_out, deg);
}